// TensorDLT_28157805592612
// MI455X (gfx1250) — compile-verified
//
#include <hip/hip_runtime.h>
#include <stdint.h>

// Truncate a generic pointer to a __shared__ object down to its 32-bit LDS
// byte offset (flat shared addresses are {aperture_hi, lds_offset32}).
__device__ __forceinline__ uint32_t lds_off_of(const void* p) {
    return (uint32_t)(uintptr_t)p;
}

// BN > 0: compile-time batch stride (lets all 8 corner loads become
// saddr+imm-offset forms in one clause). BN == 0: runtime stride fallback.
template <int BN>
__global__ __launch_bounds__(256) void TensorDLT_kernel(
    const float* __restrict__ corners,   // (8, B)
    const float* __restrict__ h4pt,      // (B, 8)
    float* __restrict__ out,             // (B, 9)
    int Bn_rt)
{
    const int Bn = (BN > 0) ? BN : Bn_rt;

    // smem: first 2048 floats = h4pt staging (32B/thread);
    // reused afterwards as 2304-float output staging (9 floats/thread).
    __shared__ __align__(16) float smem[2304];

    const int t = threadIdx.x;
    const int b = blockIdx.x * 256 + t;

    // ---- CDNA5 async path: stage this lane's 32B of h4pt into LDS ----
    {
        uint32_t so = lds_off_of(&smem[t * 8]);
        const float* gp = h4pt + (size_t)b * 8;
        // INST_OFFSET applies to both the LDS and global side -> one base, two tiles.
        asm volatile("global_load_async_to_lds_b128 %0, %1, off"
                     :: "v"(so), "v"(gp) : "memory");
        asm volatile("global_load_async_to_lds_b128 %0, %1, off offset:16"
                     :: "v"(so), "v"(gp) : "memory");
    }

    // ---- overlap: coalesced stride-B corner loads while async DMA is in flight
    float ya[4], xa[4];
#pragma unroll
    for (int i = 0; i < 4; ++i) {
        ya[i] = corners[(size_t)(2 * i)     * Bn + b];
        xa[i] = corners[(size_t)(2 * i + 1) * Bn + b];
    }

    asm volatile("s_wait_asynccnt 0" ::: "memory");   // own-lane data -> wave wait only

    float dy[4], dx[4];
    {
        const float4* sp = (const float4*)(&smem[t * 8]);
        float4 d0 = sp[0], d1 = sp[1];    // [dy0,dx0,dy1,dx1], [dy2,dx2,dy3,dx3]
        dy[0] = d0.x; dx[0] = d0.y; dy[1] = d0.z; dx[1] = d0.w;
        dy[2] = d1.x; dx[2] = d1.y; dy[3] = d1.z; dx[3] = d1.w;
    }
    __syncthreads();   // staging reads done everywhere; smem can be reused

    // ---- structured DLT solve ---------------------------------------------
    // Odd rows:  y h1 + x h2 + h3 - yb(y h7 + x h8) = yb   (group A)
    // Even rows: y h4 + x h5 + h6 - xb(y h7 + x h8) = xb   (group B)
    // Shared 4x3 left block C=[y,x,1]; augmented row = [C | A(2)+rhs | B(2)+rhs]
    float R[4][9];
#pragma unroll
    for (int i = 0; i < 4; ++i) {
        float y = ya[i], x = xa[i];
        float ybv = y + dy[i], xbv = x + dx[i];
        R[i][0] = y;        R[i][1] = x;        R[i][2] = 1.0f;
        R[i][3] = -ybv * y; R[i][4] = -ybv * x; R[i][5] = ybv;
        R[i][6] = -xbv * y; R[i][7] = -xbv * x; R[i][8] = xbv;
    }

    // Forward elimination of the 3 shared columns with partial pivoting
    // (branch-free conditional row swaps -> v_cndmask chains).
#pragma unroll
    for (int k = 0; k < 3; ++k) {
#pragma unroll
        for (int j = k + 1; j < 4; ++j) {
            bool sw = __builtin_fabsf(R[j][k]) > __builtin_fabsf(R[k][k]);
#pragma unroll
            for (int c = k; c < 9; ++c) {
                float hi = R[k][c], lo = R[j][c];
                R[k][c] = sw ? lo : hi;
                R[j][c] = sw ? hi : lo;
            }
        }
        float inv = 1.0f / R[k][k];
#pragma unroll
        for (int j = k + 1; j < 4; ++j) {
            float m = R[j][k] * inv;
#pragma unroll
            for (int c = k + 1; c < 9; ++c) R[j][c] -= m * R[k][c];
        }
    }

    // Row 3 now has zero C-part: one equation per group in (h7,h8) -> 2x2 solve
    float a0 = R[3][3], a1 = R[3][4], ra = R[3][5];
    float b0 = R[3][6], b1 = R[3][7], rb = R[3][8];
    float idet = 1.0f / (a0 * b1 - a1 * b0);
    float h7 = (ra * b1 - a1 * rb) * idet;
    float h8 = (a0 * rb - ra * b0) * idet;

    // Triangular back-substitution, both groups share the pivots of C.
    float i22 = 1.0f / R[2][2], i11 = 1.0f / R[1][1], i00 = 1.0f / R[0][0];
    float h3 = (R[2][5] - R[2][3] * h7 - R[2][4] * h8) * i22;
    float h2 = (R[1][5] - R[1][3] * h7 - R[1][4] * h8 - R[1][2] * h3) * i11;
    float h1 = (R[0][5] - R[0][3] * h7 - R[0][4] * h8 - R[0][1] * h2 - R[0][2] * h3) * i00;
    float h6 = (R[2][8] - R[2][6] * h7 - R[2][7] * h8) * i22;
    float h5 = (R[1][8] - R[1][6] * h7 - R[1][7] * h8 - R[1][2] * h6) * i11;
    float h4 = (R[0][8] - R[0][6] * h7 - R[0][7] * h8 - R[0][1] * h5 - R[0][2] * h6) * i00;

    // ---- output: stage 9 floats/thread in LDS, then 576 coalesced b128
    //      async stores straight from LDS (largest stream of the kernel).
    smem[t * 9 + 0] = h1; smem[t * 9 + 1] = h2; smem[t * 9 + 2] = h3;
    smem[t * 9 + 3] = h4; smem[t * 9 + 4] = h5; smem[t * 9 + 5] = h6;
    smem[t * 9 + 6] = h7; smem[t * 9 + 7] = h8; smem[t * 9 + 8] = 1.0f;
    __syncthreads();

    float* obase = out + (size_t)blockIdx.x * 2304;
#pragma unroll
    for (int r = 0; r < 3; ++r) {
        int idx = t + r * 256;
        if (idx < 576) {
            uint32_t soff = lds_off_of(&smem[idx * 4]);
            const float* op = obase + idx * 4;
            asm volatile("global_store_async_from_lds_b128 %0, %1, off"
                         :: "v"(op), "v"(soff) : "memory");
        }
    }
    // S_ENDPGM performs an implicit wait-idle, draining ASYNCcnt.
}

extern "C" void kernel_launch(void* const* d_in, const int* in_sizes, int n_in,
                              void* d_out, int out_size, void* d_ws, size_t ws_size,
                              hipStream_t stream) {
    (void)n_in; (void)out_size; (void)d_ws; (void)ws_size;
    const float* corners = (const float*)d_in[0];   // (8, B) float32
    const float* h4pt    = (const float*)d_in[1];   // (B, 8) float32
    float* out = (float*)d_out;                     // (B, 3, 3) float32
    int Bn = in_sizes[0] / 8;                       // B = 524288 (multiple of 256)
    int grid = Bn / 256;
    if (Bn == 524288) {
        TensorDLT_kernel<524288><<<grid, 256, 0, stream>>>(corners, h4pt, out, Bn);
    } else {
        TensorDLT_kernel<0><<<grid, 256, 0, stream>>>(corners, h4pt, out, Bn);
    }
}